// BondMessageEmbedding_47888885351052
// MI455X (gfx1250) — compile-verified
//
#include <hip/hip_runtime.h>

typedef __attribute__((ext_vector_type(2))) float v2f;
typedef __attribute__((ext_vector_type(8))) float v8f;

#define ATOM_DIM_C 128
#define BOND_DIM_C 64
#define IN_DIM_C   192     // ATOM_DIM + BOND_DIM
#define MSG_DIM_C  256
#define BLOCK_E    32      // edges per workgroup (2 M-tiles of 16)
#define KCHUNK     32      // K staged per iteration
#define A_PAD      196     // stride%64 == 4  -> conflict-free A frags, 16B rows
#define B_PAD      36      // stride%64 == 36 -> conflict-free B frags, 16B rows

// Low 32 bits of a flat pointer into __shared__ are the wave-relative LDS
// byte offset on gfx1250 (flat->LDS mapping truncates to addr[31:0]).
__device__ __forceinline__ unsigned lds_off(const void* p) {
    return (unsigned)(unsigned long long)p;
}

__device__ __forceinline__ void wait_async0() {
#if defined(__has_builtin) && __has_builtin(__builtin_amdgcn_s_wait_asynccnt)
    __builtin_amdgcn_s_wait_asynccnt(0);
#else
    asm volatile("s_wait_asynccnt 0" ::: "memory");
#endif
}

// GVS-mode async copy: 16 bytes/lane from sbase + voff + imm_off to LDS dst + imm_off.
#define ASYNC_LDS_B128(dst, voff, sbase, imm_off)                              \
    asm volatile("global_load_async_to_lds_b128 %0, %1, %2 offset:%3"          \
                 :: "v"(dst), "v"(voff), "s"(sbase), "i"(imm_off) : "memory")

__global__ __launch_bounds__(256) void bond_msg_wmma_f32(
    const float* __restrict__ V,
    const float* __restrict__ E,
    const long long* __restrict__ edge_src,   // row 0 of edge_index[2, n_edges]
    const float* __restrict__ W,              // [MSG_DIM, IN_DIM] row-major
    const float* __restrict__ bias,           // [MSG_DIM]
    float* __restrict__ out)                  // [n_edges, MSG_DIM]
{
    __shared__ float As[BLOCK_E][A_PAD];   // gathered h tile: [edge][k]
    __shared__ float Bs[MSG_DIM_C][B_PAD]; // W chunk: [n][k_local]
    __shared__ int   srcs[BLOCK_E];

    const int t    = threadIdx.x;
    const int lane = t & 31;
    const int wave = t >> 5;
    const int half = lane >> 4;   // 0: lanes 0-15, 1: lanes 16-31
    const int l16  = lane & 15;
    const long long eb = (long long)blockIdx.x * BLOCK_E;

    const unsigned long long Vb = (unsigned long long)V;
    const unsigned long long Eb = (unsigned long long)E;
    const unsigned long long Wb = (unsigned long long)W;

    if (t < BLOCK_E) srcs[t] = (int)edge_src[eb + t];
    __syncthreads();

    // ---- Async-stage gathered h = [V[src] | E] tile straight into LDS ----
    {
        const int row = t >> 3;   // 0..31 (edge within tile)
        const int l8  = t & 7;    // 8 threads per row, 6 x b128 each
        const unsigned vbase = (unsigned)srcs[row] * (ATOM_DIM_C * 4u);
        const unsigned ebase = (unsigned)(eb + row) * (BOND_DIM_C * 4u);
        #pragma unroll
        for (int i = 0; i < 6; ++i) {
            const int q = l8 + (i << 3);                  // quad index 0..47
            const unsigned dst = lds_off(&As[row][q << 2]);
            if (q < (ATOM_DIM_C / 4)) {
                const unsigned voff = vbase + (unsigned)(q << 4);
                ASYNC_LDS_B128(dst, voff, Vb, 0);
            } else {
                const unsigned voff = ebase + (unsigned)((q - ATOM_DIM_C / 4) << 4);
                ASYNC_LDS_B128(dst, voff, Eb, 0);
            }
        }
    }

    // Wave work assignment: m_tile in {0,1} (16 edges), 4 N-tiles (64 cols)
    const int m_tile  = wave & 1;
    const int n_group = wave >> 1;        // 0..3
    const int arow    = m_tile * 16 + l16;

    v8f acc[4];
    #pragma unroll
    for (int j = 0; j < 4; ++j)
        acc[j] = (v8f){0.f, 0.f, 0.f, 0.f, 0.f, 0.f, 0.f, 0.f};

    const unsigned bdst = lds_off(&Bs[t][0]);

    // ---- K loop: async-stage W chunk to LDS, then WMMA over it ----
    for (int c = 0; c < IN_DIM_C / KCHUNK; ++c) {
        if (c) __syncthreads();   // previous chunk's readers done before overwrite
        {
            // thread t owns output row n = t: copy W[t][c*32 .. c*32+31].
            // Instruction offset advances global and LDS address together.
            const unsigned woff = (unsigned)(t * IN_DIM_C + c * KCHUNK) * 4u;
            ASYNC_LDS_B128(bdst, woff, Wb, 0);
            ASYNC_LDS_B128(bdst, woff, Wb, 16);
            ASYNC_LDS_B128(bdst, woff, Wb, 32);
            ASYNC_LDS_B128(bdst, woff, Wb, 48);
            ASYNC_LDS_B128(bdst, woff, Wb, 64);
            ASYNC_LDS_B128(bdst, woff, Wb, 80);
            ASYNC_LDS_B128(bdst, woff, Wb, 96);
            ASYNC_LDS_B128(bdst, woff, Wb, 112);
        }
        wait_async0();            // covers A-tile asyncs too on c == 0
        __syncthreads();

        #pragma unroll
        for (int s = 0; s < KCHUNK / 4; ++s) {
            // A frag (16x4 f32 layout): lane holds A[m][k0], A[m][k0+1],
            // k0 = 2*half within the 4-wide K step
            const int kk = (s << 2) + (half << 1);
            v2f a = *(const v2f*)&As[arow][c * KCHUNK + kk];
            #pragma unroll
            for (int j = 0; j < 4; ++j) {
                const int nb = n_group * 64 + j * 16;
                // B frag (4x16 f32 layout): lane's column N = l16,
                // holds B[k0][N], B[k0+1][N]
                v2f bf = *(const v2f*)&Bs[nb + l16][kk];
                acc[j] = __builtin_amdgcn_wmma_f32_16x16x4_f32(
                    /*neg_a=*/false, a, /*neg_b=*/false, bf,
                    /*c_mod=*/(short)0, acc[j],
                    /*reuse_a=*/false, /*reuse_b=*/false);
            }
        }
    }

    // ---- Epilogue: bias add + store (C/D layout: VGPR r -> M = r + 8*half) ----
    #pragma unroll
    for (int j = 0; j < 4; ++j) {
        const int nb = n_group * 64 + j * 16;
        const float bv = bias[nb + l16];
        const long long row0 = eb + m_tile * 16 + half * 8;
        float* __restrict__ orow = out + row0 * MSG_DIM_C + nb + l16;
        #pragma unroll
        for (int r = 0; r < 8; ++r) {
            orow[(long long)r * MSG_DIM_C] = acc[j][r] + bv;
        }
    }
}

extern "C" void kernel_launch(void* const* d_in, const int* in_sizes, int n_in,
                              void* d_out, int out_size, void* d_ws, size_t ws_size,
                              hipStream_t stream) {
    const float*     V   = (const float*)d_in[0];
    const float*     E   = (const float*)d_in[1];
    const long long* EI  = (const long long*)d_in[2];  // int64 [2, n_edges]; row 0 = src
    const float*     W   = (const float*)d_in[3];
    const float*     b   = (const float*)d_in[4];
    float*           out = (float*)d_out;

    const int n_edges = in_sizes[1] / BOND_DIM_C;      // 800000
    dim3 grid(n_edges / BLOCK_E);
    dim3 block(256);
    bond_msg_wmma_f32<<<grid, block, 0, stream>>>(V, E, EI, W, b, out);
}